// DotPredictor_76948634075697
// MI455X (gfx1250) — compile-verified
//
#include <hip/hip_runtime.h>
#include <hip/hip_bf16.h>

// edge_dot: score[e] = <h[src[e]], h[dst[e]]>, D_FEAT = 128 fp32.
// One wave32 computes 16 edges via V_WMMA_F32_16X16X4_F32 accumulation:
//   A(16x4) = 16 src rows (K-chunk), B(4x16) = 16 dst rows (K-chunk),
//   scores = diag(sum_k A_k x B_k).
// A/B VGPR layout (ISA 7.12.2) for 32-bit 16x4 / 4x16:
//   lane L, VGPR j  ->  K-slot (j + (L>=16 ? 2 : 0)) of row (L & 15).
// Since WMMA sums over K, K-slot->feature assignment is arbitrary as long as
// A and B agree. We exploit this: each lane does ONE b128 load per matrix per
// 8 features (lanes<16 cover k0..k0+3, lanes>=16 cover k0+4..k0+7) feeding
// TWO wmma ops -> half the VMEM instruction count vs the b64 formulation.

typedef __attribute__((ext_vector_type(2))) float v2f;
typedef __attribute__((ext_vector_type(4))) float v4f;
typedef __attribute__((ext_vector_type(8))) float v8f;

#define D_FEAT 128

__global__ void __launch_bounds__(128, 1)
edge_dot_wmma_kernel(const int* __restrict__ src,
                     const int* __restrict__ dst,
                     const float* __restrict__ h,
                     float* __restrict__ out,
                     int nEdges)
{
    const int lane   = threadIdx.x & 31;
    const int waveId = threadIdx.x >> 5;
    const int wavesPerBlock = blockDim.x >> 5;
    const int e0 = (blockIdx.x * wavesPerBlock + waveId) << 4;   // 16 edges per wave

    if (e0 >= nEdges) return;   // wave-uniform branch: EXEC stays all-1s for WMMA

    // Row indices: lanes 0-15 and 16-31 both need edge (lane & 15).
    const int m = lane & 15;
    int e = e0 + m;
    if (e >= nEdges) e = nEdges - 1;             // clamp tail, keep EXEC full

    const long long srow = (long long)src[e];
    const long long drow = (long long)dst[e];

    // Column base: lanes >=16 cover the upper 4 features of each 8-feature chunk.
    const int colBase = (lane >> 4) << 2;        // 0 or 4 floats
    const float* __restrict__ ps = h + srow * D_FEAT + colBase;
    const float* __restrict__ pd = h + drow * D_FEAT + colBase;

    v8f c = {};  // 16x16 f32 accumulator (diagonal = the 16 edge scores)

#pragma unroll 4
    for (int k0 = 0; k0 < D_FEAT; k0 += 8) {
        v4f sv = *(const v4f*)(ps + k0);         // 16B aligned
        v4f dv = *(const v4f*)(pd + k0);

        v2f a0 = __builtin_shufflevector(sv, sv, 0, 1);
        v2f b0 = __builtin_shufflevector(dv, dv, 0, 1);
        v2f a1 = __builtin_shufflevector(sv, sv, 2, 3);
        v2f b1 = __builtin_shufflevector(dv, dv, 2, 3);

        // (neg_a, A, neg_b, B, c_mod, C, reuse_a, reuse_b)
        c = __builtin_amdgcn_wmma_f32_16x16x4_f32(false, a0, false, b0,
                                                  (short)0, c, false, false);
        c = __builtin_amdgcn_wmma_f32_16x16x4_f32(false, a1, false, b1,
                                                  (short)0, c, false, false);
    }

    // Diagonal extraction (C/D layout, ISA 7.12.2):
    //   edge i in 0..7  -> VGPR i,   lane i        (lanes 0-7)
    //   edge i in 8..15 -> VGPR i-8, lane i+16     (lanes 24-31)
    const int sel = lane & 7;                    // = VGPR index holding this lane's score
    float val = c[0];
    if (sel == 1) val = c[1];
    if (sel == 2) val = c[2];
    if (sel == 3) val = c[3];
    if (sel == 4) val = c[4];
    if (sel == 5) val = c[5];
    if (sel == 6) val = c[6];
    if (sel == 7) val = c[7];

    const bool writer = (lane < 8) || (lane >= 24);
    const int outIdx = e0 + ((lane < 8) ? lane : (lane - 16));
    if (writer && outIdx < nEdges) out[outIdx] = val;
}

extern "C" void kernel_launch(void* const* d_in, const int* in_sizes, int n_in,
                              void* d_out, int out_size, void* d_ws, size_t ws_size,
                              hipStream_t stream) {
    // Inputs: edge_index [2, E] int32, h [N, 128] fp32. Output: [E] fp32.
    const int E = in_sizes[0] / 2;
    const int*   src = (const int*)d_in[0];
    const int*   dst = src + E;
    const float* h   = (const float*)d_in[1];
    float*       out = (float*)d_out;

    const int wavesPerBlock = 4;                 // 128 threads
    const int threads = wavesPerBlock * 32;
    const int totalWaves = (E + 15) / 16;        // 16 edges per wave
    const int blocks = (totalWaves + wavesPerBlock - 1) / wavesPerBlock;

    edge_dot_wmma_kernel<<<blocks, threads, 0, stream>>>(src, dst, h, out, E);
}